// WLHN_19696720020228
// MI455X (gfx1250) — compile-verified
//
#include <hip/hip_runtime.h>
#include <hip/hip_bf16.h>
#include <math.h>

typedef __attribute__((ext_vector_type(2))) float v2f;
typedef __attribute__((ext_vector_type(8))) float v8f;

#define DD        64
#define DIN       128
#define NCLASS    10
#define NGRAPH    512
#define MIN_NORMF 1e-15f
#define BALL_EPSF 0.004f
#define BN_EPSF   1e-5f
#define SCALINGF  0.4621171572600098f   /* tanh(0.5) */

__device__ __forceinline__ float wave_sum32(float v) {
#pragma unroll
  for (int m = 16; m >= 1; m >>= 1) v += __shfl_xor(v, m, 32);
  return v;
}

/* ---------------- generic utilities ---------------- */

__global__ void k_zero(float* __restrict__ p, int n) {
  int i = blockIdx.x * blockDim.x + threadIdx.x;
  if (i < n) p[i] = 0.f;
}

__global__ void k_copy(float* __restrict__ dst, const float* __restrict__ src, int n) {
  int i = blockIdx.x * blockDim.x + threadIdx.x;
  if (i < n) dst[i] = src[i];
}

/* ---------------- WMMA fp32 GEMM: C = relu?(A[MxK] @ B[KxN] + bias) ----------------
   K, Ncol, relu, zerolast are compile-time so every load folds to
   saddr + scaled-voffset + immediate; M % 16 == 0 required (always true here).
   One wave computes a 16x64 stripe: A fragment (b64 load) shared across 4 n-tiles,
   4 independent V_WMMA_F32_16X16X4_F32 accumulator chains for ILP.               */

template <int K, int NCOL, bool RELU, bool ZEROLAST>
__global__ void k_gemm_wmma_f32(const float* __restrict__ A, const float* __restrict__ B,
                                const float* __restrict__ bias, float* __restrict__ C,
                                int M) {
  constexpr int WAVECOLS = NCOL >> 6;          // stripes of 4 n-tiles
  int wave = (blockIdx.x * blockDim.x + threadIdx.x) >> 5;
  int lane = threadIdx.x & 31;
  int tilesM = M >> 4;
  if (wave >= tilesM * WAVECOLS) return;
  int tm = wave / WAVECOLS, wc = wave % WAVECOLS;
  int mBase = tm << 4;
  int nBase = wc << 6;
  int half = lane >> 4, l16 = lane & 15;

  int aOff = (mBase + l16) * K + (half << 1);        // element index into A
  int bOff = nBase + l16 + (half << 1) * NCOL;       // element index into B

  v8f acc0 = {}, acc1 = {}, acc2 = {}, acc3 = {};
#pragma unroll 4
  for (int kb = 0; kb < K; kb += 4) {
    v2f a = *reinterpret_cast<const v2f*>(A + (aOff + kb));   // global_load_b64
    int bi = bOff + kb * NCOL;                                // NCOL compile-time
    v2f b0 = { B[bi],      B[bi + NCOL]      };
    v2f b1 = { B[bi + 16], B[bi + NCOL + 16] };
    v2f b2 = { B[bi + 32], B[bi + NCOL + 32] };
    v2f b3 = { B[bi + 48], B[bi + NCOL + 48] };
    acc0 = __builtin_amdgcn_wmma_f32_16x16x4_f32(false, a, false, b0, (short)0, acc0, false, false);
    acc1 = __builtin_amdgcn_wmma_f32_16x16x4_f32(false, a, false, b1, (short)0, acc1, false, false);
    acc2 = __builtin_amdgcn_wmma_f32_16x16x4_f32(false, a, false, b2, (short)0, acc2, false, false);
    acc3 = __builtin_amdgcn_wmma_f32_16x16x4_f32(false, a, false, b3, (short)0, acc3, false, false);
  }

#pragma unroll
  for (int t = 0; t < 4; ++t) {
    const v8f& acc = (t == 0) ? acc0 : (t == 1) ? acc1 : (t == 2) ? acc2 : acc3;
    int col = nBase + (t << 4) + l16;
    float bv = bias[col];
#pragma unroll
    for (int j = 0; j < 8; ++j) {
      int row = mBase + j + (half << 3);   // VGPR j: lanes0-15 M=j, lanes16-31 M=8+j
      float v = acc[j] + bv;
      if (RELU) v = fmaxf(v, 0.f);
      if (ZEROLAST && col == NCOL - 1) v = 0.f;
      C[row * NCOL + col] = v;
    }
  }
}

/* ---------------- edge scatter-add: H[dst] += X[src] ---------------- */

__global__ void k_edge_scatter(const int* __restrict__ ei, const float* __restrict__ X,
                               float* __restrict__ H, int E) {
  int t = blockIdx.x * blockDim.x + threadIdx.x;
  if (t >= E * 16) return;
  int e = t >> 4, cb = (t & 15) << 2;
  int s = ei[e], d = ei[E + e];
  const float4 v = *reinterpret_cast<const float4*>(&X[(size_t)s * DD + cb]);
  atomicAdd(&H[(size_t)d * DD + cb + 0], v.x);
  atomicAdd(&H[(size_t)d * DD + cb + 1], v.y);
  atomicAdd(&H[(size_t)d * DD + cb + 2], v.z);
  atomicAdd(&H[(size_t)d * DD + cb + 3], v.w);
}

/* ---------------- batchnorm ---------------- */

__global__ void k_bn_stats(const float* __restrict__ H, float* __restrict__ stats, int n) {
  __shared__ float s1[256], s2[256];
  int c = threadIdx.x & 63;
  int g = threadIdx.x >> 6;  // 4 row-groups per block
  float a = 0.f, b = 0.f;
  for (int r = blockIdx.x * 4 + g; r < n; r += gridDim.x * 4) {
    float v = H[(size_t)r * DD + c];
    a += v; b += v * v;
  }
  s1[threadIdx.x] = a; s2[threadIdx.x] = b;
  __syncthreads();
  if (threadIdx.x < 128) { s1[threadIdx.x] += s1[threadIdx.x + 128]; s2[threadIdx.x] += s2[threadIdx.x + 128]; }
  __syncthreads();
  if (threadIdx.x < 64) {
    atomicAdd(&stats[threadIdx.x],      s1[threadIdx.x] + s1[threadIdx.x + 64]);
    atomicAdd(&stats[64 + threadIdx.x], s2[threadIdx.x] + s2[threadIdx.x + 64]);
  }
}

__global__ void k_bn_apply(float* __restrict__ H, const float* __restrict__ stats,
                           const float* __restrict__ gamma, const float* __restrict__ beta, int n) {
  int i = blockIdx.x * blockDim.x + threadIdx.x;
  if (i >= n * DD) return;
  int c = i & 63;
  float inv_n = 1.f / (float)n;
  float mu  = stats[c] * inv_n;
  float var = stats[64 + c] * inv_n - mu * mu;
  H[i] = gamma[c] * (H[i] - mu) * rsqrtf(var + BN_EPSF) + beta[c];
}

/* ---------------- hyperbolic z-chain (one wave32 per node, 2 ch/lane) ---------------- */

__global__ void k_z_init(const float* __restrict__ X, float* __restrict__ Z, int n) {
  int node = (blockIdx.x * blockDim.x + threadIdx.x) >> 5;
  int lane = threadIdx.x & 31;
  if (node >= n) return;
  size_t b = (size_t)node * DD + lane * 2;
  float x0 = X[b], x1 = X[b + 1];
  float nrm = fmaxf(sqrtf(wave_sum32(x0 * x0 + x1 * x1)), MIN_NORMF);
  float maxn = 1.f - BALL_EPSF;
  float sc = ((nrm > maxn) ? (maxn / nrm) : 1.f) * SCALINGF;
  Z[b] = x0 * sc; Z[b + 1] = x1 * sc;
}

__global__ void k_z_update(const float* __restrict__ X, const float* __restrict__ ZP,
                           const float* __restrict__ ZC, float* __restrict__ ZN, int n) {
  int node = (blockIdx.x * blockDim.x + threadIdx.x) >> 5;
  int lane = threadIdx.x & 31;
  if (node >= n) return;
  size_t b = (size_t)node * DD + lane * 2;
  float x0 = X[b], x1 = X[b + 1];
  float zp0 = ZP[b], zp1 = ZP[b + 1];
  float zc0 = ZC[b], zc1 = ZC[b + 1];
  /* z_children = SCALING * project(x) */
  float nx = fmaxf(sqrtf(wave_sum32(x0 * x0 + x1 * x1)), MIN_NORMF);
  float maxn = 1.f - BALL_EPSF;
  float sc = ((nx > maxn) ? (maxn / nx) : 1.f) * SCALINGF;
  float ch0 = x0 * sc, ch1 = x1 * sc;
  /* a = zcur / max(|zcur|^2, eps); r2 = |a|^2 - 1 */
  float s   = wave_sum32(zc0 * zc0 + zc1 * zc1);
  float mu2 = fmaxf(s, MIN_NORMF);
  float a0 = zc0 / mu2, a1 = zc1 / mu2;
  float r2 = s / (mu2 * mu2) - 1.f;
  /* z_parent = reflect_at_zero(zp, zcur) */
  float u0 = zp0 - a0, u1 = zp1 - a1;
  float du = fmaxf(wave_sum32(u0 * u0 + u1 * u1), MIN_NORMF);
  float f = r2 / du;
  float p0 = f * u0 + a0, p1 = f * u1 + a1;
  /* reflect_through_zero(p, e_{63}, ch) */
  float pn = fmaxf(sqrtf(wave_sum32(p0 * p0 + p1 * p1)), MIN_NORMF);
  float ph0 = p0 / pn, ph1 = p1 / pn;
  float q1 = (lane == 31) ? 1.f : 0.f;        /* p_vec = e_63 lives in lane31 ch1 */
  float rr0 = -ph0, rr1 = q1 - ph1;
  float m = wave_sum32(rr0 * ch0 + rr1 * ch1) / wave_sum32(rr0 * rr0 + rr1 * rr1);
  float cc0 = ch0 - 2.f * rr0 * m, cc1 = ch1 - 2.f * rr1 * m;
  /* reflect_at_zero(cc, zcur) */
  float v0 = cc0 - a0, v1 = cc1 - a1;
  float dv = fmaxf(wave_sum32(v0 * v0 + v1 * v1), MIN_NORMF);
  float g = r2 / dv;
  ZN[b] = g * v0 + a0; ZN[b + 1] = g * v1 + a1;
}

__global__ void k_pool_logmap(const float* __restrict__ Z, const int* __restrict__ batch,
                              float* __restrict__ pooled, int n) {
  int node = (blockIdx.x * blockDim.x + threadIdx.x) >> 5;
  int lane = threadIdx.x & 31;
  if (node >= n) return;
  size_t b = (size_t)node * DD + lane * 2;
  float z0 = Z[b], z1 = Z[b + 1];
  float yn = fmaxf(sqrtf(wave_sum32(z0 * z0 + z1 * z1)), MIN_NORMF);
  float t = fminf(fmaxf(yn, -1.f + 1e-15f), 1.f - 1e-15f);
  float sc = atanhf(t) / yn;
  int g = batch[node];
  atomicAdd(&pooled[(size_t)g * DD + lane * 2],     z0 * sc);
  atomicAdd(&pooled[(size_t)g * DD + lane * 2 + 1], z1 * sc);
}

/* ---------------- fc3 + log_softmax ---------------- */

__global__ void k_head(const float* __restrict__ T2, const float* __restrict__ W,
                       const float* __restrict__ bv, float* __restrict__ out) {
  int row = blockIdx.x * blockDim.x + threadIdx.x;
  if (row >= NGRAPH) return;
  float logits[NCLASS];
#pragma unroll
  for (int c = 0; c < NCLASS; ++c) {
    float acc = bv[c];
    for (int k = 0; k < DD; ++k) acc += T2[(size_t)row * DD + k] * W[k * NCLASS + c];
    logits[c] = acc;
  }
  float mx = logits[0];
#pragma unroll
  for (int c = 1; c < NCLASS; ++c) mx = fmaxf(mx, logits[c]);
  float sum = 0.f;
#pragma unroll
  for (int c = 0; c < NCLASS; ++c) sum += expf(logits[c] - mx);
  float lse = mx + logf(sum);
#pragma unroll
  for (int c = 0; c < NCLASS; ++c) out[(size_t)row * NCLASS + c] = logits[c] - lse;
}

/* ---------------- host orchestration ---------------- */

static inline int gemm_blocks(int M, int Ncol) {
  int tiles = (M / 16) * (Ncol / 64);          // one wave per 16x64 stripe
  return (tiles * 32 + 255) / 256;
}

extern "C" void kernel_launch(void* const* d_in, const int* in_sizes, int n_in,
                              void* d_out, int out_size, void* d_ws, size_t ws_size,
                              hipStream_t stream) {
  const float* x     = (const float*)d_in[0];
  const int*   ei    = (const int*)  d_in[1];
  const int*   batch = (const int*)  d_in[2];
  const float* W0    = (const float*)d_in[3];
  const float* b0    = (const float*)d_in[4];
  const float* cW1   = (const float*)d_in[5];
  const float* cb1   = (const float*)d_in[6];
  const float* gamma = (const float*)d_in[7];
  const float* beta  = (const float*)d_in[8];
  const float* cW2   = (const float*)d_in[9];
  const float* cb2   = (const float*)d_in[10];
  const float* f1W   = (const float*)d_in[11];
  const float* f1b   = (const float*)d_in[12];
  const float* f2W   = (const float*)d_in[13];
  const float* f2b   = (const float*)d_in[14];
  const float* f3W   = (const float*)d_in[15];
  const float* f3b   = (const float*)d_in[16];
  float* out = (float*)d_out;

  const int N = in_sizes[0] / DIN;     // 50000
  const int E = in_sizes[1] / 2;       // 800000
  const size_t NF = (size_t)N * DD;

  float* ws = (float*)d_ws;
  float* X      = ws;
  float* H      = X + NF;
  float* H2     = H + NF;
  float* ZA     = H2 + NF;
  float* ZB     = ZA + NF;
  float* ZC     = ZB + NF;
  float* pooled = ZC + NF;                       // 512*64
  float* T1     = pooled + (size_t)NGRAPH * DD;  // 512*128
  float* T2     = T1 + (size_t)NGRAPH * 128;     // 512*64
  float* stats  = T2 + (size_t)NGRAPH * DD;      // 128

  const int elemBlocks = (int)((NF + 255) / 256);
  const int nodeBlocks = (N * 32 + 255) / 256;

  /* x0 = relu(x @ W0 + b0) */
  k_gemm_wmma_f32<DIN, DD, true, false>
      <<<gemm_blocks(N, DD), 256, 0, stream>>>(x, W0, b0, X, N);

  /* z[0] = 0, z[1] = SCALING*project(x0)  (unique_rows collapses to identity) */
  k_zero<<<elemBlocks, 256, 0, stream>>>(ZA, (int)NF);
  k_z_init<<<nodeBlocks, 256, 0, stream>>>(X, ZB, N);

  float* zp = ZA; float* zc = ZB; float* zn = ZC;
  for (int i = 0; i < 3; ++i) {
    /* h = x + segment_sum(x[src], dst) */
    k_copy<<<elemBlocks, 256, 0, stream>>>(H, X, (int)NF);
    k_edge_scatter<<<(E * 16 + 255) / 256, 256, 0, stream>>>(ei, X, H, E);
    /* h = relu(h @ convW1 + b1); batchnorm */
    k_gemm_wmma_f32<DD, DD, true, false>
        <<<gemm_blocks(N, DD), 256, 0, stream>>>(H, cW1 + (size_t)i * DD * DD,
                                                 cb1 + (size_t)i * DD, H2, N);
    k_zero<<<1, 128, 0, stream>>>(stats, 128);
    k_bn_stats<<<256, 256, 0, stream>>>(H2, stats, N);
    k_bn_apply<<<elemBlocks, 256, 0, stream>>>(H2, stats, gamma + (size_t)i * DD,
                                               beta + (size_t)i * DD, N);
    /* x = relu(h @ convW2 + b2); x[:, -1] = 0 */
    k_gemm_wmma_f32<DD, DD, true, true>
        <<<gemm_blocks(N, DD), 256, 0, stream>>>(H2, cW2 + (size_t)i * DD * DD,
                                                 cb2 + (size_t)i * DD, X, N);
    /* hyperbolic chain */
    k_z_update<<<nodeBlocks, 256, 0, stream>>>(X, zp, zc, zn, N);
    float* t = zp; zp = zc; zc = zn; zn = t;
  }

  /* pooled = segment_sum(logmap0(z[-1]), batch) */
  k_zero<<<(NGRAPH * DD + 255) / 256, 256, 0, stream>>>(pooled, NGRAPH * DD);
  k_pool_logmap<<<nodeBlocks, 256, 0, stream>>>(zc, batch, pooled, N);

  /* FC head */
  k_gemm_wmma_f32<DD, 128, true, false>
      <<<gemm_blocks(NGRAPH, 128), 256, 0, stream>>>(pooled, f1W, f1b, T1, NGRAPH);
  k_gemm_wmma_f32<128, DD, true, false>
      <<<gemm_blocks(NGRAPH, DD), 256, 0, stream>>>(T1, f2W, f2b, T2, NGRAPH);
  k_head<<<(NGRAPH + 255) / 256, 256, 0, stream>>>(T2, f3W, f3b, out);
}